// TypeLayer_56349970923980
// MI455X (gfx1250) — compile-verified
//
#include <hip/hip_runtime.h>
#include <hip/hip_bf16.h>

// ---------------------------------------------------------------------------
// TypeLayer for MI455X (gfx1250, wave32).
//
// Reference:  fact_val = rel_features[rels] @ W.T + b      [NF=1e6, D=128]
//             out = relu(seg_sum(fact_val, tails) + seg_sum(fact_val, heads))
//
// Key transform: only NR=2000 distinct relations -> precompute
//   rel_out = rel_features @ W.T + b  (2000x128, fp32 WMMA, exact vs ref)
// then scatter rel_out[rels[f]] into ONE accumulator for both heads and
// tails (relu(a+b) of two segment sums == relu of the combined scatter).
//
// Roofline: scatter dominates. 2M (fact,dst) row-updates x 128 f32 atomics
// = 256M GLOBAL_ATOMIC_ADD_F32 into an 8MB accumulator resident in the
// 192MB L2; HBM traffic ~25MB (~1us at 23.3TB/s) -> L2-atomic-throughput
// bound. Hence: non-returning *hardware* fadd atomics (unsafeAtomicAdd, no
// CAS-loop fallback), combined accumulator (halves atomic count), float4
// coalesced gathers of the hot rel_out table.
// ---------------------------------------------------------------------------

typedef __attribute__((ext_vector_type(2))) float v2f;
typedef __attribute__((ext_vector_type(8))) float v8f;

#define D_DIM 128

// ---------------------------------------------------------------------------
// Kernel 1: rel_out[r, n] = sum_k rel_features[r, k] * W[n, k] + b[n]
// One wave32 per 16x16 output tile using V_WMMA_F32_16X16X4_F32 (fp32 exact
// vs reference; dense part is only 65 MFLOP so no reason to drop precision).
//
// A-frag layout (ISA 7.12.2, 32-bit A 16x4): lanes 0-15 hold M=lane, K=0 (v0)
// and K=1 (v1); lanes 16-31 hold M=lane-16, K=2 (v0) and K=3 (v1).
// B (4x16) mirrors with N=lane&15 since B[k][n] = W[n][k].
// C/D layout: VGPR v, lanes 0-15 -> M=v, lanes 16-31 -> M=v+8, N=lane&15.
// ---------------------------------------------------------------------------
__global__ __launch_bounds__(256) void relout_gemm_wmma(
    const float* __restrict__ relf,   // [NR, 128]
    const float* __restrict__ W,      // [128, 128]
    const float* __restrict__ bias,   // [128]
    float* __restrict__ rel_out,      // [NR, 128]
    int nr)
{
    const int lane  = threadIdx.x & 31;
    const int wave  = threadIdx.x >> 5;   // 0..7  -> N tile
    const int mtile = blockIdx.x;         // 0..nr/16-1

    const int row16 = lane & 15;
    const int khalf = (lane >> 4) << 1;   // 0 or 2

    const float* __restrict__ arow = relf + (size_t)(mtile * 16 + row16) * D_DIM;
    const float* __restrict__ brow = W    + (size_t)(wave  * 16 + row16) * D_DIM;

    v8f c = {};
#pragma unroll 4
    for (int k0 = 0; k0 < D_DIM; k0 += 4) {
        v2f a, b;
        a.x = arow[k0 + khalf];
        a.y = arow[k0 + khalf + 1];
        b.x = brow[k0 + khalf];
        b.y = brow[k0 + khalf + 1];
        // 8 args: (neg_a, A, neg_b, B, c_mod, C, reuse_a, reuse_b)
        c = __builtin_amdgcn_wmma_f32_16x16x4_f32(
                false, a, false, b, (short)0, c, false, false);
    }

    const int mbase = mtile * 16 + ((lane >> 4) << 3);  // +8 for upper lanes
    const int col   = wave * 16 + (lane & 15);
    const float bv  = bias[col];
    float* __restrict__ outp = rel_out + (size_t)mbase * D_DIM + col;
#pragma unroll
    for (int v = 0; v < 8; ++v)
        outp[(size_t)v * D_DIM] = c[v] + bv;
}

// ---------------------------------------------------------------------------
// Kernel 2: scatter. One wave32 per fact; each lane handles 4 columns
// (float4 gather of the rel_out row = one coalesced 512B wave read), then
// 4+4 non-returning hardware f32 fadd atomics into the combined accumulator.
// Indices are loaded once per wave (lane 0) and broadcast.
// ---------------------------------------------------------------------------
__global__ __launch_bounds__(256) void scatter_facts(
    const int* __restrict__ rels,
    const int* __restrict__ heads,
    const int* __restrict__ tails,
    const float* __restrict__ rel_out,  // [NR, 128]
    float* __restrict__ acc,            // [B*E, 128] (zeroed)
    int nf)
{
    const long long tid = (long long)blockIdx.x * blockDim.x + threadIdx.x;
    const int f = (int)(tid >> 5);
    if (f >= nf) return;
    const int lane = (int)tid & 31;
    const int c4   = lane << 2;

    // Wave-uniform index fetch: lane 0 loads, broadcast to the wave.
    int r = 0, h = 0, t = 0;
    if (lane == 0) {
        r = rels[f];
        h = heads[f];
        t = tails[f];
    }
    r = __shfl(r, 0);
    h = __shfl(h, 0);
    t = __shfl(t, 0);

    const float4 v = *(const float4*)(rel_out + (size_t)r * D_DIM + c4);

    float* pt = acc + (size_t)t * D_DIM + c4;
    float* ph = acc + (size_t)h * D_DIM + c4;

    // unsafeAtomicAdd -> GLOBAL_ATOMIC_ADD_F32 (no CAS-loop expansion),
    // non-returning form (cheapest atomic path through L2).
    unsafeAtomicAdd(pt + 0, v.x); unsafeAtomicAdd(pt + 1, v.y);
    unsafeAtomicAdd(pt + 2, v.z); unsafeAtomicAdd(pt + 3, v.w);
    unsafeAtomicAdd(ph + 0, v.x); unsafeAtomicAdd(ph + 1, v.y);
    unsafeAtomicAdd(ph + 2, v.z); unsafeAtomicAdd(ph + 3, v.w);
}

// ---------------------------------------------------------------------------
// Kernel 3: out = relu(acc), vectorized float4.
// ---------------------------------------------------------------------------
__global__ __launch_bounds__(256) void relu_out_kernel(
    const float* __restrict__ acc, float* __restrict__ out, int n4)
{
    const int i = blockIdx.x * blockDim.x + threadIdx.x;
    if (i >= n4) return;
    float4 v = ((const float4*)acc)[i];
    v.x = fmaxf(v.x, 0.f);
    v.y = fmaxf(v.y, 0.f);
    v.z = fmaxf(v.z, 0.f);
    v.w = fmaxf(v.w, 0.f);
    ((float4*)out)[i] = v;
}

// ---------------------------------------------------------------------------
// Inputs (setup_inputs order):
//  0: local_entity [B,E] int   (unused; shape only)
//  1: heads        [NF]  int
//  2: rels         [NF]  int
//  3: tails        [NF]  int
//  4: rel_features [NR,128] f32
//  5: W            [128,128] f32
//  6: b            [128] f32
// Output: [B,E,128] f32
// ---------------------------------------------------------------------------
extern "C" void kernel_launch(void* const* d_in, const int* in_sizes, int n_in,
                              void* d_out, int out_size, void* d_ws, size_t ws_size,
                              hipStream_t stream)
{
    const int*   heads = (const int*)d_in[1];
    const int*   rels  = (const int*)d_in[2];
    const int*   tails = (const int*)d_in[3];
    const float* relf  = (const float*)d_in[4];
    const float* W     = (const float*)d_in[5];
    const float* bias  = (const float*)d_in[6];
    float*       out   = (float*)d_out;

    const int nf   = in_sizes[1];            // 1,000,000
    const int nr   = in_sizes[4] / D_DIM;    // 2000
    const int nseg = out_size / D_DIM;       // B*E = 16000

    // Workspace layout: [rel_out: nr*128 f32][acc: nseg*128 f32]
    float* rel_out = (float*)d_ws;
    float* acc     = rel_out + (size_t)nr * D_DIM;

    // Accumulator must be zeroed every call (deterministic, graph-safe).
    hipMemsetAsync(acc, 0, (size_t)nseg * D_DIM * sizeof(float), stream);

    // 1) Tiny dense GEMM on the WMMA path (fp32, exact vs reference).
    relout_gemm_wmma<<<nr / 16, 256, 0, stream>>>(relf, W, bias, rel_out, nr);

    // 2) Atomic scatter of both head and tail contributions.
    const long long total = (long long)nf * 32;
    const int blocks = (int)((total + 255) / 256);
    scatter_facts<<<blocks, 256, 0, stream>>>(rels, heads, tails, rel_out, acc, nf);

    // 3) Fused relu + store to output.
    const int n4 = out_size / 4;
    relu_out_kernel<<<(n4 + 255) / 256, 256, 0, stream>>>(acc, out, n4);
}